// CapsuleLayer_9526237462669
// MI455X (gfx1250) — compile-verified
//
#include <hip/hip_runtime.h>
#include <cfloat>
#include <cmath>

typedef __attribute__((ext_vector_type(2))) float v2f;
typedef __attribute__((ext_vector_type(8))) float v8f;
typedef __attribute__((ext_vector_type(4))) unsigned int u32x4;
typedef __attribute__((ext_vector_type(8))) int i32x8;
typedef __attribute__((ext_vector_type(4))) int i32x4;

#define T0N  8
#define T1N  8
#define Z1N  64
#define CIN  32
#define H0   64
#define W0D  64
#define H1   32
#define W1D  32
#define POSN (H1 * W1D)      // 1024
#define COUT 512             // T1N * Z1N
#define KKN  288             // CIN * 3 * 3
#define GN   64              // N * t0
#define NB   8               // batch N
#define HALO_W 33            // staged halo columns per row
#define HALO_ELEMS (CIN * 3 * HALO_W)   // 3168 floats

// ---------------------------------------------------------------------------
// Kernel 1: transpose weights [COUT][KKN] -> [KKN][COUT] for coalesced B reads
// ---------------------------------------------------------------------------
__global__ void transpose_w_kernel(const float* __restrict__ Wsrc,
                                   float* __restrict__ Wt) {
    int idx = blockIdx.x * 256 + threadIdx.x;
    if (idx < COUT * KKN) {
        int c = idx / KKN;
        int k = idx - c * KKN;
        Wt[k * COUT + c] = Wsrc[idx];
    }
}

// ---------------------------------------------------------------------------
// Kernel 2: zero routing logits
// ---------------------------------------------------------------------------
__global__ void zero_kernel(float* __restrict__ p, int n) {
    int idx = blockIdx.x * 256 + threadIdx.x;
    if (idx < n) p[idx] = 0.0f;
}

// ---------------------------------------------------------------------------
// Kernel 3: implicit-GEMM conv via V_WMMA_F32_16X16X4_F32 (fp32, K=4)
//   grid = (posTiles=64, g=64), block = 256 (8 waves of 32)
//   Each block: 1 image, 16 contiguous positions (half a row), all 512 chans.
//   Input halo (32ch x 3rows x 33cols) staged in LDS:
//     - interior tiles: one TDM tensor_load_to_lds (3D tile descriptor)
//     - border tiles  : manual per-lane loads with zero fill
//   u_hat layout: [g][pos][COUT], channel-innermost.
// ---------------------------------------------------------------------------
__global__ __launch_bounds__(256) void conv_wmma_kernel(
    const float* __restrict__ x, const float* __restrict__ Wt,
    const float* __restrict__ bias, float* __restrict__ uhat) {
    __shared__ float sm[HALO_ELEMS];   // [ch][row][33], tight (TDM-compatible)
    __shared__ int offT[KKN];          // k -> LDS base offset of (ch,kh,kw)

    const int tileIdx = blockIdx.x;       // 0..63 position tile
    const int g       = blockIdx.y;       // 0..63 image
    const int pos0    = tileIdx * 16;
    const int h1      = pos0 >> 5;        // output row (tiles never cross rows)
    const int wb      = pos0 & 31;        // 0 or 16

    // k -> im2col LDS offset table (kills div/mod in the WMMA loop)
    for (int k = threadIdx.x; k < KKN; k += 256) {
        int ch = k / 9;
        int r  = k - ch * 9;
        int kh = r / 3;
        int kw = r - kh * 3;
        offT[k] = (ch * 3 + kh) * HALO_W + kw;
    }

    const float* xg = x + (size_t)g * CIN * H0 * W0D;
    const int h0s = 2 * h1 - 1;   // first input row of the halo
    const int w0s = 2 * wb - 1;   // first input col of the halo

    // interior tile: entire 33x3x32 halo is inside the image -> TDM fast path
    const bool interior = (h0s >= 0) && (w0s >= 0);  // high edges always fit

    if (interior) {
        if (threadIdx.x < 32) {   // wave-uniform: exactly one wave issues TDM
            const unsigned lds_base =
                (unsigned)(size_t)(&sm[0]);                 // LDS byte offset
            const unsigned long long ga =
                (unsigned long long)(size_t)(xg + h0s * W0D + w0s);

            u32x4 g0;
            g0.x = 1u;                                      // count=1 (valid D#)
            g0.y = lds_base;                                // lds_addr
            g0.z = (unsigned)(ga & 0xffffffffu);            // global_addr[31:0]
            g0.w = (unsigned)((ga >> 32) & 0x01ffffffu)     // global_addr[56:32]
                   | (2u << 30);                            // type=2 (image)

            i32x8 g1;
            g1[0] = (2 << 16);            // wg_mask=0, data_size=2 (4 bytes)
            g1[1] = (int)(64u << 16);     // tensor_dim0[15:0]=64 (row width)
            g1[2] = (int)(64u << 16);     // tensor_dim1[15:0]=64 (rows)
            g1[3] = (int)((unsigned)HALO_W << 16);  // tile_dim0=33
            g1[4] = (int)(3u | (32u << 16));        // tile_dim1=3, tile_dim2=32
            g1[5] = 64;                   // tensor_dim0_stride=64  (row pitch)
            g1[6] = (int)(4096u << 16);   // tensor_dim1_stride[15:0]=4096 (chan)
            g1[7] = 0;                    // tensor_dim1_stride[47:16]=0

            i32x4 g2;
            g2[0] = 32;                   // tensor_dim2 = 32 channels
            g2[1] = 0; g2[2] = 0; g2[3] = 0;
            i32x4 g3;
            g3[0] = 0; g3[1] = 0; g3[2] = 0; g3[3] = 0;
            i32x8 g4;                     // extra group (clang-23 6-arg form)
            g4[0] = 0; g4[1] = 0; g4[2] = 0; g4[3] = 0;
            g4[4] = 0; g4[5] = 0; g4[6] = 0; g4[7] = 0;

            __builtin_amdgcn_tensor_load_to_lds(g0, g1, g2, g3, g4, 0);
            __builtin_amdgcn_s_wait_tensorcnt(0);
        }
    } else {
        // border tile: manual load with zero fill outside the image
        for (int idx = threadIdx.x; idx < HALO_ELEMS; idx += 256) {
            int ch  = idx / (3 * HALO_W);
            int rem = idx - ch * (3 * HALO_W);
            int rr  = rem / HALO_W;
            int cc  = rem - rr * HALO_W;
            int h0  = h0s + rr;
            int w0  = w0s + cc;
            float v = 0.0f;
            if (h0 >= 0 && h0 < H0 && w0 >= 0 && w0 < W0D)
                v = xg[ch * (H0 * W0D) + h0 * W0D + w0];
            sm[(ch * 3 + rr) * HALO_W + cc] = v;
        }
    }
    __syncthreads();

    const int lane = threadIdx.x & 31;
    const int wave = threadIdx.x >> 5;          // 0..7
    const int m    = lane & 15;                 // M index (position in tile)
    const int ksel = (lane >> 4) << 1;          // K offset 0 or 2 (f32 16x4 layout)
    const int nIdx = lane & 15;                 // N index (channel in tile)
    const float* smm = sm + 2 * m;

    v8f acc[4];
#pragma unroll
    for (int i = 0; i < 4; i++) {
        v8f z = {0.f, 0.f, 0.f, 0.f, 0.f, 0.f, 0.f, 0.f};
        acc[i] = z;
    }

#pragma unroll 2
    for (int k0 = 0; k0 < KKN; k0 += 4) {
        const int ka = k0 + ksel;
        // A fragment (im2col from LDS): A[m][k], k = {ka, ka+1} per lane half
        v2f a;
        a.x = smm[offT[ka]];
        a.y = smm[offT[ka + 1]];

        const float* wka = Wt + (size_t)ka * COUT + nIdx;
        __builtin_prefetch(wka + 4 * COUT, 0, 0);   // next K-chunk of weights

#pragma unroll
        for (int i = 0; i < 4; i++) {
            const int c0 = wave * 16 + i * 128;
            v2f bf;
            bf.x = wka[c0];
            bf.y = wka[c0 + COUT];
            // D = A(16x4 f32) x B(4x16 f32) + C(16x16 f32)
            acc[i] = __builtin_amdgcn_wmma_f32_16x16x4_f32(
                false, a, false, bf, (short)0, acc[i], false, false);
        }
    }

    // store D: lane(0-15)->N=lane, VGPR v -> M=v (+8 for lanes 16-31); + bias
    const int mAdd = (lane >> 4) << 3;   // 0 or 8
#pragma unroll
    for (int i = 0; i < 4; i++) {
        const int c0 = wave * 16 + i * 128;
        const int ch = c0 + nIdx;
        const float bv = bias[ch];
#pragma unroll
        for (int v = 0; v < 8; v++) {
            const int posI = pos0 + v + mAdd;
            uhat[((size_t)g * POSN + posI) * COUT + ch] = acc[i][v] + bv;
        }
    }
}

// ---------------------------------------------------------------------------
// Kernel 4: one routing iteration. grid = N*POSN blocks, block = 256.
//   r = softmax_t1(maxpool3x3(b)); p = sum_t0 r*u_hat; v = squash(p);
//   if !last: b += sum_z1 u_hat*v ; else write v to out.
// ---------------------------------------------------------------------------
__global__ __launch_bounds__(256) void routing_kernel(
    const float* __restrict__ uhat, float* __restrict__ bbuf,
    float* __restrict__ out, int last) {
    __shared__ float sm_m[64];      // maxpooled logits per (t0,t1)
    __shared__ float sm_r[64];      // routing weights
    __shared__ float sm_p[512];     // p per (t1,z1)
    __shared__ float sm_n2[8];      // squared norm per t1
    __shared__ float sm_bacc[64];   // agreement accumulators

    const int t   = threadIdx.x;
    const int blk = blockIdx.x;             // 0..8191
    const int n   = blk >> 10;
    const int pos = blk & 1023;
    const int h   = pos >> 5, w = pos & 31;

    if (t < 64) sm_bacc[t] = 0.0f;

    // 1) maxpool 3x3 (stride 1, pad 1) of b
    if (t < 64) {
        const int t0 = t >> 3, t1 = t & 7;
        const float* bp = bbuf + ((size_t)((n * T0N + t0) * T1N + t1)) * POSN;
        float mx = -FLT_MAX;
        for (int dh = -1; dh <= 1; dh++) {
            int hh = h + dh;
            if (hh < 0 || hh >= H1) continue;
            for (int dw = -1; dw <= 1; dw++) {
                int ww = w + dw;
                if (ww < 0 || ww >= W1D) continue;
                mx = fmaxf(mx, bp[hh * W1D + ww]);
            }
        }
        sm_m[t] = mx;
    }
    __syncthreads();

    // 2) softmax over t1 (per t0)
    if (t < 64) {
        const int t0 = t >> 3;
        float mx = -FLT_MAX;
        for (int j = 0; j < 8; j++) mx = fmaxf(mx, sm_m[t0 * 8 + j]);
        float s = 0.0f;
        for (int j = 0; j < 8; j++) s += expf(sm_m[t0 * 8 + j] - mx);
        sm_r[t] = expf(sm_m[t] - mx) / s;
    }
    __syncthreads();

    // 3) p = sum_t0 r * u_hat  (each thread owns channels c1 and c1+256)
    const int c1 = t, c2 = t + 256;
    const int t1a = c1 >> 6, t1b = c2 >> 6;
    float uh1[8], uh2[8];
    float p1 = 0.0f, p2 = 0.0f;
#pragma unroll
    for (int t0 = 0; t0 < 8; t0++) {
        const float* up = uhat + ((size_t)((n * T0N + t0) * POSN + pos)) * COUT;
        const float u1 = up[c1];
        const float u2 = up[c2];
        uh1[t0] = u1;
        uh2[t0] = u2;
        p1 += sm_r[t0 * 8 + t1a] * u1;
        p2 += sm_r[t0 * 8 + t1b] * u2;
    }
    sm_p[c1] = p1;
    sm_p[c2] = p2;
    __syncthreads();

    // 4) squared norms over z1
    if (t < 8) {
        float s = 0.0f;
        for (int z = 0; z < 64; z++) {
            float q = sm_p[t * 64 + z];
            s += q * q;
        }
        sm_n2[t] = s;
    }
    __syncthreads();

    // 5) squash
    const float n2a = sm_n2[t1a], n2b = sm_n2[t1b];
    const float sa = n2a / (1.0f + n2a) / sqrtf(n2a + 1e-9f);
    const float sb = n2b / (1.0f + n2b) / sqrtf(n2b + 1e-9f);
    const float v1 = sa * p1, v2 = sb * p2;

    if (last) {
        // out[((n*8+t1)*64+z1)*1024 + pos] == out[(n*512+c)*1024 + pos]
        out[((size_t)(n * COUT + c1)) * POSN + pos] = v1;
        out[((size_t)(n * COUT + c2)) * POSN + pos] = v2;
    } else {
        // 6) agreement: b[t0,t1] += sum_z1 u_hat * v   (LDS ds_add_f32)
#pragma unroll
        for (int t0 = 0; t0 < 8; t0++) {
            atomicAdd(&sm_bacc[t0 * 8 + t1a], uh1[t0] * v1);
            atomicAdd(&sm_bacc[t0 * 8 + t1b], uh2[t0] * v2);
        }
        __syncthreads();
        if (t < 64) {
            const int t0 = t >> 3, t1 = t & 7;
            bbuf[((size_t)((n * T0N + t0) * T1N + t1)) * POSN + pos] += sm_bacc[t];
        }
    }
}

// ---------------------------------------------------------------------------
extern "C" void kernel_launch(void* const* d_in, const int* in_sizes, int n_in,
                              void* d_out, int out_size, void* d_ws,
                              size_t ws_size, hipStream_t stream) {
    const float* x    = (const float*)d_in[0];  // [8,8,32,64,64]
    const float* W    = (const float*)d_in[1];  // [512,32,3,3]
    const float* bias = (const float*)d_in[2];  // [512]
    float* out = (float*)d_out;                 // [8,8,64,32,32]

    // workspace layout
    float* uhat = (float*)d_ws;                          // 64*1024*512 floats
    float* Wt   = uhat + (size_t)GN * POSN * COUT;       // 288*512 floats
    float* bbuf = Wt + (size_t)KKN * COUT;               // 8*8*8*1024 floats
    const int bElems = NB * T0N * T1N * POSN;

    transpose_w_kernel<<<(COUT * KKN + 255) / 256, 256, 0, stream>>>(W, Wt);
    zero_kernel<<<(bElems + 255) / 256, 256, 0, stream>>>(bbuf, bElems);
    conv_wmma_kernel<<<dim3(64, 64), 256, 0, stream>>>(x, Wt, bias, uhat);

    for (int d = 0; d < 3; d++) {
        routing_kernel<<<NB * POSN, 256, 0, stream>>>(uhat, bbuf, out,
                                                      (d == 2) ? 1 : 0);
    }
}